// ModelNew_50379966382563
// MI455X (gfx1250) — compile-verified
//
#include <hip/hip_runtime.h>

typedef __attribute__((ext_vector_type(2))) float v2f;
typedef __attribute__((ext_vector_type(8))) float v8f;

namespace {
constexpr int kDFeat        = 64;   // feature dim (floats per row)
constexpr int kEdgesPerTile = 16;   // one 16x16 WMMA tile of edges per wave
}

// out[e] = dot(src_feat[src_idx[e]], dst_feat[dst_idx[e]])
// Per 16-edge tile: out_tile = diag(S * D^T) via V_WMMA_F32_16X16X4_F32,
// accumulating 16 chunks of K=4 over the 64-float feature dim.
__global__ __launch_bounds__(256) void sddmm_wmma_f32_kernel(
    const int*   __restrict__ src_idx,
    const int*   __restrict__ dst_idx,
    const float* __restrict__ src_feat,
    const float* __restrict__ dst_feat,
    float*       __restrict__ out,
    int n_edges)
{
  const int lane    = threadIdx.x & 31;
  const int wave    = (int)((blockIdx.x * (unsigned)blockDim.x + threadIdx.x) >> 5);
  const int n_tiles = (n_edges + kEdgesPerTile - 1) / kEdgesPerTile;
  if (wave >= n_tiles) return;  // wave-uniform: EXEC stays all-1s for WMMA

  // A-matrix 16x4 f32 layout (ISA 7.12.2):
  //   lanes 0-15 : M=lane,    VGPR0=K0, VGPR1=K1
  //   lanes 16-31: M=lane-16, VGPR0=K2, VGPR1=K3
  // B = D^T mirrors this, so both gathers share one offset formula:
  //   rowptr + chunk*4 + (lane>>4)*2   (floats)
  const int row   = lane & 15;     // edge within tile (M and N)
  const int khalf = lane >> 4;     // 0 -> K{0,1}, 1 -> K{2,3}

  int e = wave * kEdgesPerTile + row;
  const int e_clamped = e < n_edges ? e : (n_edges - 1);  // tail-safe gather

  const int si = src_idx[e_clamped];
  const int di = dst_idx[e_clamped];

  const float* srow = src_feat + (size_t)si * kDFeat + khalf * 2;
  const float* drow = dst_feat + (size_t)di * kDFeat + khalf * 2;

  v8f acc = {};  // 16x16 f32 C/D accumulator (8 VGPRs)
#pragma unroll
  for (int c = 0; c < kDFeat / 4; ++c) {
    v2f a = *(const v2f*)(srow + c * 4);   // S[row][4c + 2*khalf .. +1]
    v2f b = *(const v2f*)(drow + c * 4);   // D[row][4c + 2*khalf .. +1]  (== B^T layout)
    // 8 args: (neg_a, A, neg_b, B, c_mod, C, reuse_a, reuse_b)
    acc = __builtin_amdgcn_wmma_f32_16x16x4_f32(
        /*neg_a=*/false, a, /*neg_b=*/false, b,
        /*c_mod=*/(short)0, acc, /*reuse_a=*/false, /*reuse_b=*/false);
  }

  // Diagonal of C (16x16 f32 layout):
  //   m in [0,8)  -> acc[m]   at lane m
  //   m in [8,16) -> acc[m-8] at lane m+16  (lanes 24..31)
  float dval = 0.0f;
#pragma unroll
  for (int r = 0; r < 8; ++r)
    dval = ((lane & 7) == r) ? acc[r] : dval;

  int m = -1;
  if (lane < 8)        m = lane;        // diag 0..7
  else if (lane >= 24) m = lane - 16;   // diag 8..15
  if (m >= 0) {
    const int ge = wave * kEdgesPerTile + m;
    if (ge < n_edges) out[ge] = dval;
  }
}

extern "C" void kernel_launch(void* const* d_in, const int* in_sizes, int n_in,
                              void* d_out, int out_size, void* d_ws, size_t ws_size,
                              hipStream_t stream) {
  const int*   src_idx  = (const int*)  d_in[0];
  const int*   dst_idx  = (const int*)  d_in[1];
  const float* src_feat = (const float*)d_in[2];
  const float* dst_feat = (const float*)d_in[3];
  float*       out      = (float*)d_out;

  const int n_edges = in_sizes[0];                       // 1,250,000
  const int n_tiles = (n_edges + kEdgesPerTile - 1) / kEdgesPerTile;  // 78125
  const int threads = 256;                               // 8 wave32s per block
  const int waves_per_block = threads / 32;
  const int blocks  = (n_tiles + waves_per_block - 1) / waves_per_block;

  sddmm_wmma_f32_kernel<<<blocks, threads, 0, stream>>>(
      src_idx, dst_idx, src_feat, dst_feat, out, n_edges);
}